// GIN_79688823210542
// MI455X (gfx1250) — compile-verified
//
#include <hip/hip_runtime.h>

// ---------------------------------------------------------------------------
// GIN on MI455X (gfx1250, wave32).
//   Once per launch: repack weights into WMMA B-fragment order; build CSR of
//   the edge list (int atomics only).
//   Per layer:  t   = MLP_l(h)              (WMMA f32 16x16x4, fused MLP)
//               agg = gather-reduce(t, CSR) (atomic-free, one write per row)
//               h   = MLP_l(h + agg)        (same WMMA kernel, in-place)
//   Final:      out = h @ Wf + bf
// N = 100000 (= 6250*16), D = 64, E = 1200000, L = 3. Everything L2-resident.
// ---------------------------------------------------------------------------

typedef __attribute__((ext_vector_type(2))) float v2f;
typedef __attribute__((ext_vector_type(8))) float v8f;

#define NDIM 64
#define WAVES_PER_BLOCK 4
#define WP_ELEMS 4096  // packed 64x64 weight matrix: 16 chunks * 4 tiles * 32 lanes * 2
#define SCAN_B 256

__device__ __forceinline__ v8f wmma_f32_16x16x4(v2f a, v2f b, v8f c) {
  return __builtin_amdgcn_wmma_f32_16x16x4_f32(false, a, false, b, (short)0, c,
                                               false, false);
}

// Repack W (64x64 row-major) into per-WMMA B fragments:
// Wp[((c*4 + n)*32 + lane)] = { W[(4c+2*hi)  ][16n+laneM],
//                               W[(4c+2*hi)+1][16n+laneM] }
__global__ void gin_repack_kernel(const float* __restrict__ W,
                                  float* __restrict__ Wp) {
  const int tid = blockIdx.x * blockDim.x + threadIdx.x;  // 0..2047
  const int c = tid >> 7;
  const int n = (tid >> 5) & 3;
  const int lane = tid & 31;
  const int hi = lane >> 4;
  const int m = lane & 15;
  const int k = 4 * c + 2 * hi;
  Wp[tid * 2 + 0] = W[(k + 0) * NDIM + 16 * n + m];
  Wp[tid * 2 + 1] = W[(k + 1) * NDIM + 16 * n + m];
}

// MODE 0: OUT = relu(H @ W1 + b1) @ W2 + b2            (messages)
// MODE 1: OUT = relu((H+AGG) @ W1 + b1) @ W2 + b2      (combine; OUT may == H)
// MODE 2: OUT = H @ W1 + b1                            (final linear)
template <int MODE>
__global__ void __launch_bounds__(WAVES_PER_BLOCK * 32)
gin_mlp_kernel(const float* __restrict__ H, const float* __restrict__ AGG,
               const float* __restrict__ WP1, const float* __restrict__ B1,
               const float* __restrict__ WP2, const float* __restrict__ B2,
               float* __restrict__ OUT, int nTiles) {
  __shared__ float lds[WAVES_PER_BLOCK][16 * NDIM];  // wave-private 16x64 tile

  const int wave = threadIdx.x >> 5;
  const int lane = threadIdx.x & 31;
  const int tile = blockIdx.x * WAVES_PER_BLOCK + wave;
  if (tile >= nTiles) return;  // wave-uniform: EXEC all-ones for WMMA

  const int laneM = lane & 15;
  const int hi = lane >> 4;
  const int row0 = tile * 16;

  // ---- A fragments of Z = H (+ AGG) ----------------------------------------
  v2f a[16];
  {
    const size_t base = (size_t)(row0 + laneM) * NDIM + 2 * hi;
    const float* hp = H + base;
    const float* gp = AGG + base;  // dereferenced only when MODE == 1
#pragma unroll
    for (int c = 0; c < 16; ++c) {
      v2f z = *(const v2f*)(hp + 4 * c);
      if (MODE == 1) {
        v2f g = *(const v2f*)(gp + 4 * c);
        z.x += g.x;
        z.y += g.y;
      }
      a[c] = z;
    }
  }

  const v2f* wp1 = (const v2f*)WP1;

  // ---- GEMM1 ---------------------------------------------------------------
  v8f acc[4];
#pragma unroll
  for (int n = 0; n < 4; ++n) {
    const float bv = B1[16 * n + laneM];
#pragma unroll
    for (int r = 0; r < 8; ++r) acc[n][r] = bv;
  }
#pragma unroll
  for (int c = 0; c < 16; ++c)
#pragma unroll
    for (int n = 0; n < 4; ++n)
      acc[n] = wmma_f32_16x16x4(a[c], wp1[(c * 4 + n) * 32 + lane], acc[n]);

  if (MODE == 2) {
#pragma unroll
    for (int n = 0; n < 4; ++n)
#pragma unroll
      for (int r = 0; r < 8; ++r)
        OUT[(size_t)(row0 + r + 8 * hi) * NDIM + 16 * n + laneM] = acc[n][r];
    return;
  }

  // ---- ReLU, C-layout -> A-layout via wave-private LDS tile ----------------
  float* lt = lds[wave];
#pragma unroll
  for (int n = 0; n < 4; ++n)
#pragma unroll
    for (int r = 0; r < 8; ++r)
      lt[(r + 8 * hi) * NDIM + 16 * n + laneM] = fmaxf(acc[n][r], 0.0f);

  // LDS ops from one wave are in-order; same-array aliasing preserves order.
#pragma unroll
  for (int c = 0; c < 16; ++c)
    a[c] = *(const v2f*)(lt + laneM * NDIM + 4 * c + 2 * hi);

  // ---- GEMM2 ---------------------------------------------------------------
  const v2f* wp2 = (const v2f*)WP2;
#pragma unroll
  for (int n = 0; n < 4; ++n) {
    const float bv = B2[16 * n + laneM];
#pragma unroll
    for (int r = 0; r < 8; ++r) acc[n][r] = bv;
  }
#pragma unroll
  for (int c = 0; c < 16; ++c)
#pragma unroll
    for (int n = 0; n < 4; ++n)
      acc[n] = wmma_f32_16x16x4(a[c], wp2[(c * 4 + n) * 32 + lane], acc[n]);

#pragma unroll
  for (int n = 0; n < 4; ++n)
#pragma unroll
    for (int r = 0; r < 8; ++r)
      OUT[(size_t)(row0 + r + 8 * hi) * NDIM + 16 * n + laneM] = acc[n][r];
}

// ---------------- CSR construction (once per launch) ------------------------

__global__ void gin_zero_i32_kernel(int* __restrict__ p, int n) {
  const int i = blockIdx.x * blockDim.x + threadIdx.x;
  if (i < n) p[i] = 0;
}

__global__ void gin_degree_kernel(const int* __restrict__ rowp,
                                  int* __restrict__ deg, int E) {
  const int e = blockIdx.x * blockDim.x + threadIdx.x;
  if (e < E) atomicAdd(&deg[rowp[e]], 1);
}

// Per-block inclusive scan of deg into rowstart[i+1]; block sums to bwork.
__global__ void gin_scan1_kernel(const int* __restrict__ deg,
                                 int* __restrict__ rowstart1,
                                 int* __restrict__ bwork, int n) {
  __shared__ int s[SCAN_B];
  const int i = blockIdx.x * SCAN_B + threadIdx.x;
  const int v = (i < n) ? deg[i] : 0;
  s[threadIdx.x] = v;
  __syncthreads();
#pragma unroll
  for (int off = 1; off < SCAN_B; off <<= 1) {
    const int tv = (threadIdx.x >= off) ? s[threadIdx.x - off] : 0;
    __syncthreads();
    s[threadIdx.x] += tv;
    __syncthreads();
  }
  if (i < n) rowstart1[i] = s[threadIdx.x];
  if (threadIdx.x == SCAN_B - 1) bwork[blockIdx.x] = s[SCAN_B - 1];
}

// Single-block exclusive scan of block sums (nb <= 512), in place.
__global__ void gin_scan2_kernel(int* __restrict__ bwork, int nb) {
  __shared__ int s[512];
  const int i = threadIdx.x;
  const int v = (i < nb) ? bwork[i] : 0;
  s[i] = v;
  __syncthreads();
#pragma unroll
  for (int off = 1; off < 512; off <<= 1) {
    const int tv = (i >= off) ? s[i - off] : 0;
    __syncthreads();
    s[i] += tv;
    __syncthreads();
  }
  if (i < nb) bwork[i] = s[i] - v;  // exclusive offset
}

__global__ void gin_scan3_kernel(int* __restrict__ rowstart,
                                 const int* __restrict__ bwork, int n) {
  const int i = blockIdx.x * blockDim.x + threadIdx.x;
  if (i < n) rowstart[i + 1] += bwork[i / SCAN_B];
  if (i == 0) rowstart[0] = 0;
}

__global__ void gin_fill_kernel(const int* __restrict__ rowp,
                                const int* __restrict__ colp,
                                const int* __restrict__ rowstart,
                                int* __restrict__ cursor,
                                int* __restrict__ perm, int E) {
  const int e = blockIdx.x * blockDim.x + threadIdx.x;
  if (e < E) {
    const int r = rowp[e];
    const int pos = rowstart[r] + atomicAdd(&cursor[r], 1);
    perm[pos] = colp[e];
  }
}

// agg[v][:] = sum_{j in [rowstart[v], rowstart[v+1])} t[perm[j]][:]
// One node per 16-lane group, float4 per lane. Atomic-free; one write per row.
__global__ void gin_gather_kernel(const float* __restrict__ T,
                                  const int* __restrict__ rowstart,
                                  const int* __restrict__ perm,
                                  float* __restrict__ AGG, int N) {
  const int g = (blockIdx.x * blockDim.x + threadIdx.x) >> 4;  // node
  if (g >= N) return;
  const int l4 = (threadIdx.x & 15) * 4;
  const int s = rowstart[g];
  const int e = rowstart[g + 1];
  float4 acc = make_float4(0.f, 0.f, 0.f, 0.f);
  for (int j = s; j < e; ++j) {
    const int c = perm[j];
    const float4 v = *(const float4*)(T + (size_t)c * NDIM + l4);
    acc.x += v.x;
    acc.y += v.y;
    acc.z += v.z;
    acc.w += v.w;
  }
  *(float4*)(AGG + (size_t)g * NDIM + l4) = acc;
}

// ---------------------------------------------------------------------------

extern "C" void kernel_launch(void* const* d_in, const int* in_sizes, int n_in,
                              void* d_out, int out_size, void* d_ws,
                              size_t ws_size, hipStream_t stream) {
  (void)n_in;
  (void)out_size;
  (void)ws_size;

  const float* x = (const float*)d_in[0];
  const int* ei = (const int*)d_in[1];
  const float* W1 = (const float*)d_in[2];
  const float* b1 = (const float*)d_in[3];
  const float* W2 = (const float*)d_in[4];
  const float* b2 = (const float*)d_in[5];
  const float* Wf = (const float*)d_in[6];
  const float* bf = (const float*)d_in[7];
  float* out = (float*)d_out;

  const int N = in_sizes[0] / NDIM;  // 100000 (multiple of 16)
  const int E = in_sizes[1] / 2;     // 1200000
  const int L = in_sizes[3] / NDIM;  // 3
  const int nTiles = N / 16;
  const size_t NB = (size_t)N * NDIM;

  // Workspace layout
  float* agg = (float*)d_ws;
  float* t = agg + NB;
  float* h = t + NB;
  float* wp = h + NB;  // 7 packed 64x64 matrices: W1[0..2], W2[0..2], Wf
  int* deg = (int*)(wp + 7 * WP_ELEMS);
  int* cursor = deg + N;
  int* rowstart = cursor + N;       // N+1 entries
  int* bwork = rowstart + (N + 1);  // scan block sums (<= 512)
  int* perm = bwork + 1024;         // E entries

  const int* rowp = ei;      // edge_index[0]
  const int* colp = ei + E;  // edge_index[1]

  // ---- Repack all weights into WMMA B-fragment order -----------------------
  const float* wsrc[7] = {W1, W1 + WP_ELEMS, W1 + 2 * WP_ELEMS,
                          W2, W2 + WP_ELEMS, W2 + 2 * WP_ELEMS, Wf};
  for (int m = 0; m < 7; ++m)
    gin_repack_kernel<<<WP_ELEMS / 2 / 256, 256, 0, stream>>>(
        wsrc[m], wp + (size_t)m * WP_ELEMS);

  // ---- Build CSR (int atomics only) ----------------------------------------
  const int nScanBlocks = (N + SCAN_B - 1) / SCAN_B;  // 391 <= 512
  gin_zero_i32_kernel<<<(2 * N + 255) / 256, 256, 0, stream>>>(deg, 2 * N);
  gin_degree_kernel<<<(E + 255) / 256, 256, 0, stream>>>(rowp, deg, E);
  gin_scan1_kernel<<<nScanBlocks, SCAN_B, 0, stream>>>(deg, rowstart + 1,
                                                       bwork, N);
  gin_scan2_kernel<<<1, 512, 0, stream>>>(bwork, nScanBlocks);
  gin_scan3_kernel<<<(N + 255) / 256, 256, 0, stream>>>(rowstart, bwork, N);
  gin_fill_kernel<<<(E + 255) / 256, 256, 0, stream>>>(rowp, colp, rowstart,
                                                       cursor, perm, E);

  // ---- Layers --------------------------------------------------------------
  const dim3 mlpBlk(WAVES_PER_BLOCK * 32);
  const dim3 mlpGrd((nTiles + WAVES_PER_BLOCK - 1) / WAVES_PER_BLOCK);
  const dim3 gatGrd((N * 16 + 255) / 256);

  const float* hcur = x;
  for (int l = 0; l < L; ++l) {
    const float* wp1 = wp + (size_t)l * WP_ELEMS;
    const float* wp2 = wp + (size_t)(3 + l) * WP_ELEMS;
    const float* v1 = b1 + (size_t)l * NDIM;
    const float* v2 = b2 + (size_t)l * NDIM;

    // t = MLP_l(h)  (MLP commutes with the edge gather)
    gin_mlp_kernel<0><<<mlpGrd, mlpBlk, 0, stream>>>(hcur, nullptr, wp1, v1,
                                                     wp2, v2, t, nTiles);
    // agg = segment_sum(t[col], row) via CSR gather (atomic-free)
    gin_gather_kernel<<<gatGrd, 256, 0, stream>>>(t, rowstart, perm, agg, N);
    // h = MLP_l(h + agg)  (in-place on h after layer 0)
    gin_mlp_kernel<1><<<mlpGrd, mlpBlk, 0, stream>>>(hcur, agg, wp1, v1, wp2,
                                                     v2, h, nTiles);
    hcur = h;
  }
  // out = h @ Wf + bf
  gin_mlp_kernel<2><<<mlpGrd, mlpBlk, 0, stream>>>(
      hcur, nullptr, wp + (size_t)6 * WP_ELEMS, bf, nullptr, nullptr, out,
      nTiles);
}